// GCNEncoder_42769284334214
// MI455X (gfx1250) — compile-verified
//
#include <hip/hip_runtime.h>

#define NNODES 50000
#define NEDGES 600000
#define HID 128

typedef float v2f __attribute__((ext_vector_type(2)));
typedef float v4f __attribute__((ext_vector_type(4)));
typedef float v8f __attribute__((ext_vector_type(8)));

// ---------------- degree / norm kernels ----------------
__global__ void k_deg_init(float* __restrict__ deg, int n) {
    int i = blockIdx.x * blockDim.x + threadIdx.x;
    if (i < n) deg[i] = 1.0f;                      // self loop
}

__global__ void k_deg_accum(const int* __restrict__ dst, float* __restrict__ deg, int e) {
    int i = blockIdx.x * blockDim.x + threadIdx.x;
    if (i < e) atomicAdd(&deg[dst[i]], 1.0f);
}

__global__ void k_deg_rsqrt(float* __restrict__ deg, int n) {
    int i = blockIdx.x * blockDim.x + threadIdx.x;
    if (i < n) {
        float d = deg[i];
        deg[i] = (d > 0.0f) ? __frsqrt_rn(d) : 0.0f;   // becomes dinv
    }
}

// ---------------- fused GEMM (h@W) + bias + self-loop init ----------------
// One 256-thread block = 8 waves; wave w computes rows [ (bid*8+w)*16 , +16 ).
// W (128x128 f32, 64KB) staged in LDS. Each wave holds its full A strip in
// registers (32 x v2f frags) and emits 8 column tiles of 16x16 via
// v_wmma_f32_16x16x4_f32, K accumulated over 32 steps.
//
// f32 WMMA VGPR layouts (ISA 7.12.2):
//   A 16x4 : lane L (r = L&15, hi = L>>4): v0 = A[r][4s+2*hi], v1 = A[r][4s+2*hi+1]
//   B 4x16 : lane L (n = L&15, hi = L>>4): v0 = B[4s+2*hi][n], v1 = B[4s+2*hi+1][n]
//   C 16x16: lane L, vgpr v -> row 8*hi+v, col L&15
template <bool RELU_IN>
__global__ __launch_bounds__(256) void k_gcn_gemm(
    const float* __restrict__ A,     // [n,128] layer input
    const float* __restrict__ W,     // [128,128]
    const float* __restrict__ bias,  // [128]
    const float* __restrict__ dinv,  // [n]
    float* __restrict__ HW,          // [n,128]  h@W
    float* __restrict__ AGG,         // [n,128]  bias + dinv^2 * hw (scatter base)
    int n)
{
    __shared__ float lw[HID * HID];  // 64 KB

    const int tid = threadIdx.x;
    // cooperative W -> LDS, b128 loads/stores
    {
        const v4f* gsrc = (const v4f*)W;
        v4f* ldst = (v4f*)lw;
        #pragma unroll
        for (int i = 0; i < (HID * HID / 4) / 256; ++i)
            ldst[tid + i * 256] = gsrc[tid + i * 256];
    }
    __syncthreads();

    const int wave = tid >> 5;
    const int lane = tid & 31;
    const int row0 = (blockIdx.x * 8 + wave) * 16;
    if (row0 >= n) return;                     // wave-uniform: EXEC stays full

    const int hi = lane >> 4;                  // 0/1
    const int r  = lane & 15;

    // ---- load A strip into registers (row r of this tile, my K half-pairs)
    v2f afrag[32];
    const float* arow = A + (size_t)(row0 + r) * HID + 2 * hi;
    #pragma unroll
    for (int s = 0; s < 32; ++s) {
        v2f a = *(const v2f*)(arow + 4 * s);
        if (RELU_IN) { a.x = fmaxf(a.x, 0.0f); a.y = fmaxf(a.y, 0.0f); }
        afrag[s] = a;
    }

    // self-loop norms for the 8 rows this lane will store (row = row0+8*hi+v)
    float sn[8];
    #pragma unroll
    for (int v = 0; v < 8; ++v) {
        float dv = dinv[row0 + 8 * hi + v];
        sn[v] = dv * dv;
    }

    for (int t = 0; t < 8; ++t) {              // 8 column tiles
        const int n0 = t * 16;
        v8f c = {0.f, 0.f, 0.f, 0.f, 0.f, 0.f, 0.f, 0.f};
        #pragma unroll
        for (int s = 0; s < 32; ++s) {
            v2f b;
            b.x = lw[(4 * s + 2 * hi + 0) * HID + n0 + r];
            b.y = lw[(4 * s + 2 * hi + 1) * HID + n0 + r];
            c = __builtin_amdgcn_wmma_f32_16x16x4_f32(
                    false, afrag[s], false, b, (short)0, c, false, false);
        }
        const float bv = bias[n0 + r];
        #pragma unroll
        for (int v = 0; v < 8; ++v) {
            const size_t m = (size_t)(row0 + 8 * hi + v);
            const float hwv = c[v];
            HW [m * HID + n0 + r] = hwv;
            AGG[m * HID + n0 + r] = bv + hwv * sn[v];
        }
    }
}

// ---------------- edge scatter: agg[dst] += norm * hw[src] ----------------
// One wave per edge; each lane moves 4 floats (float4 gather, 4 f32 atomics).
__global__ __launch_bounds__(256) void k_gcn_scatter(
    const int* __restrict__ src, const int* __restrict__ dst,
    const float* __restrict__ dinv,
    const float* __restrict__ HW, float* __restrict__ AGG, int e)
{
    const int gwave = (blockIdx.x * 256 + threadIdx.x) >> 5;
    const int lane  = threadIdx.x & 31;
    if (gwave >= e) return;

    const int s = src[gwave];
    const int d = dst[gwave];
    const float norm = dinv[s] * dinv[d];

    const v4f hv = *(const v4f*)(HW + (size_t)s * HID + lane * 4);
    float* out = AGG + (size_t)d * HID + lane * 4;
    atomicAdd(out + 0, hv.x * norm);
    atomicAdd(out + 1, hv.y * norm);
    atomicAdd(out + 2, hv.z * norm);
    atomicAdd(out + 3, hv.w * norm);
}

// ---------------- final ReLU ----------------
__global__ void k_relu(const float* __restrict__ in, float* __restrict__ out, int n) {
    int i = blockIdx.x * blockDim.x + threadIdx.x;
    if (i < n) out[i] = fmaxf(in[i], 0.0f);
}

// ---------------- launcher ----------------
extern "C" void kernel_launch(void* const* d_in, const int* in_sizes, int n_in,
                              void* d_out, int out_size, void* d_ws, size_t ws_size,
                              hipStream_t stream)
{
    const float* x  = (const float*)d_in[0];
    const int*   ei = (const int*)d_in[1];     // [2, E] : row0 = src, row1 = dst
    const float* W0 = (const float*)d_in[2];
    const float* b0 = (const float*)d_in[3];
    const float* W1 = (const float*)d_in[4];
    const float* b1 = (const float*)d_in[5];
    const float* W2 = (const float*)d_in[6];
    const float* b2 = (const float*)d_in[7];

    const int n = NNODES, e = NEDGES;
    const int* srcp = ei;
    const int* dstp = ei + e;

    float* hw   = (float*)d_ws;                      // [n,128]
    float* bufA = hw + (size_t)n * HID;              // [n,128]
    float* dinv = bufA + (size_t)n * HID;            // [n]  (deg -> dinv in place)
    float* bufB = (float*)d_out;                     // [n,128] reuse output buffer

    // normalization
    k_deg_init <<<(n + 255) / 256, 256, 0, stream>>>(dinv, n);
    k_deg_accum<<<(e + 255) / 256, 256, 0, stream>>>(dstp, dinv, e);
    k_deg_rsqrt<<<(n + 255) / 256, 256, 0, stream>>>(dinv, n);

    const int gemmGrid = (n + 127) / 128;            // 128 rows / block
    const int scatGrid = (e + 7) / 8;                // 8 edges (waves) / block

    // layer 1: x -> bufA
    k_gcn_gemm<false><<<gemmGrid, 256, 0, stream>>>(x,    W0, b0, dinv, hw, bufA, n);
    k_gcn_scatter    <<<scatGrid, 256, 0, stream>>>(srcp, dstp, dinv, hw, bufA, e);
    // layer 2: relu(bufA) -> bufB (= d_out storage)
    k_gcn_gemm<true> <<<gemmGrid, 256, 0, stream>>>(bufA, W1, b1, dinv, hw, bufB, n);
    k_gcn_scatter    <<<scatGrid, 256, 0, stream>>>(srcp, dstp, dinv, hw, bufB, e);
    // layer 3: relu(bufB) -> bufA
    k_gcn_gemm<true> <<<gemmGrid, 256, 0, stream>>>(bufB, W2, b2, dinv, hw, bufA, n);
    k_gcn_scatter    <<<scatGrid, 256, 0, stream>>>(srcp, dstp, dinv, hw, bufA, e);
    // output = relu(bufA)
    k_relu<<<((size_t)n * HID + 255) / 256, 256, 0, stream>>>(bufA, (float*)d_out, n * HID);
}